// Attention_24283745092194
// MI455X (gfx1250) — compile-verified
//
#include <hip/hip_runtime.h>

// Problem constants (from reference)
#define NB     4      // batch
#define DIM    512
#define SEQ    2048
#define NHEADS 8
#define HDIM   32
#define INNER  256    // NHEADS*HDIM
#define SCALE  0.17677669529663687f   // HDIM^-0.5

typedef __attribute__((ext_vector_type(2))) float v2f;
typedef __attribute__((ext_vector_type(4))) float v4f;
typedef __attribute__((ext_vector_type(8))) float v8f;

// D(16x16 f32) = A(16x4 f32) x B(4x16 f32) + C   -- native f32 WMMA on CDNA5
__device__ __forceinline__ v8f wmma4(v2f a, v2f b, v8f c) {
  return __builtin_amdgcn_wmma_f32_16x16x4_f32(
      /*neg_a=*/false, a, /*neg_b=*/false, b,
      /*c_mod=*/(short)0, c, /*reuse_a=*/false, /*reuse_b=*/false);
}

// ---------------------------------------------------------------------------
// Kernel 1: QKV projection,  qkv[b,o,n] = sum_c w_qkv[o,c] * x[b,c,n]
// Orientation: M = o (rows), N = n (cols)  ->
//   A = w_qkv tile (contiguous b64 loads), B = x tile (2 scalar loads/step,
//   coalesced across lanes), D rows = o: q/k stores contiguous in d (b128),
//   v written TRANSPOSED as (B,H,d,N) for the attention kernel's A operand.
// Block = 256 thr (8 waves) per (b, n-tile); wave covers 96 output channels.
// ---------------------------------------------------------------------------
__global__ void __launch_bounds__(256)
qkv_kernel(const float* __restrict__ x, const float* __restrict__ wqkv,
           float* __restrict__ q, float* __restrict__ k, float* __restrict__ vT) {
  const int lane = threadIdx.x & 31;
  const int wave = threadIdx.x >> 5;
  const int b    = blockIdx.x >> 7;          // SEQ/16 = 128 n-tiles
  const int n0   = (blockIdx.x & 127) * 16;
  const int mcol = lane & 15;
  const bool hi  = lane >= 16;
  const int kb   = hi ? 2 : 0;
  const float* xb = x + (size_t)b * DIM * SEQ;
  const int o_base = wave * 96;

  v8f acc[6];
  v8f z = {0.f,0.f,0.f,0.f,0.f,0.f,0.f,0.f};
  #pragma unroll
  for (int t = 0; t < 6; ++t) acc[t] = z;

  for (int c0 = 0; c0 < DIM; c0 += 4) {
    v2f bx;                                  // B operand from x (K x n)
    const float* xp = xb + (size_t)(c0 + kb) * SEQ + n0 + mcol;
    bx.x = xp[0];
    bx.y = xp[SEQ];
    #pragma unroll
    for (int t = 0; t < 6; ++t) {
      const int o = o_base + t * 16 + mcol;  // A row per lane
      v2f a = *(const v2f*)(wqkv + (size_t)o * DIM + c0 + kb);
      acc[t] = wmma4(a, bx, acc[t]);
    }
  }

  const int n = n0 + mcol;                   // D column per lane
  #pragma unroll
  for (int t = 0; t < 6; ++t) {
    const int o0t = o_base + t * 16;         // tile-uniform
    const int wh  = o0t >> 8;                // 0=q 1=k 2=v
    const int oi0 = o0t & 255;
    const int h   = oi0 >> 5;
    const int ddb = (oi0 & 31) + (hi ? 8 : 0);  // first of 8 consecutive d's
    if (wh < 2) {
      float* dst = (wh == 0) ? q : k;
      float* p = dst + (((size_t)b * NHEADS + h) * SEQ + n) * HDIM + ddb;
      v4f* pa = (v4f*)&acc[t];
      *(v4f*)p       = pa[0];
      *(v4f*)(p + 4) = pa[1];
    } else {
      // vT[b][h][dd][n], dd = ddb + r  (stride SEQ scatter, 8 stores)
      float* p = vT + (((size_t)b * NHEADS + h) * HDIM + ddb) * SEQ + n;
      #pragma unroll
      for (int r = 0; r < 8; ++r) p[(size_t)r * SEQ] = acc[t][r];
    }
  }
}

// ---------------------------------------------------------------------------
// Kernel 2: squared L2 norms of q and k rows.
// ---------------------------------------------------------------------------
__global__ void __launch_bounds__(256)
norm_kernel(const float* __restrict__ q, const float* __restrict__ k,
            float* __restrict__ q2, float* __restrict__ k2) {
  const int i = blockIdx.x * 256 + threadIdx.x;   // (b,h,n) flat
  const float* qp = q + (size_t)i * HDIM;
  const float* kp = k + (size_t)i * HDIM;
  float sq = 0.f, sk = 0.f;
  #pragma unroll
  for (int d = 0; d < HDIM; ++d) {
    float a = qp[d]; sq += a * a;
    float c = kp[d]; sk += c * c;
  }
  q2[i] = sq; k2[i] = sk;
}

// ---------------------------------------------------------------------------
// Kernel 3: transposed flash attention.
// Wave handles 16 queries. S^T(16key x 16query) = K_tile x Q^T so that each
// query column lives in one lane: softmax reduce = 8 VGPRs + one shfl_xor(16),
// and the online-softmax rescale of O^T is a uniform per-lane multiply.
// O^T += V^T x P^T;  V^T A-operand is contiguous b64 thanks to the (H,d,N)
// v layout; P^T D-layout -> B-layout via half-wave shfl_xor(16) swaps.
// Output att stored as (B, N, INNER): contiguous b128 stores, and contiguous
// A-operand for the final projection.
// ---------------------------------------------------------------------------
__global__ void __launch_bounds__(256)
attn_kernel(const float* __restrict__ q, const float* __restrict__ k,
            const float* __restrict__ vT, const float* __restrict__ q2,
            const float* __restrict__ k2, float* __restrict__ att) {
  const int lane = threadIdx.x & 31;
  const int wave = threadIdx.x >> 5;
  const int bh    = blockIdx.x >> 4;          // 16 chunks of 128 queries
  const int chunk = blockIdx.x & 15;
  const int m0    = chunk * 128 + wave * 16;
  const int mcol  = lane & 15;
  const bool hi   = lane >= 16;
  const int kb    = hi ? 2 : 0;

  const float* qb  = q  + (size_t)bh * SEQ * HDIM;
  const float* kbp = k  + (size_t)bh * SEQ * HDIM;
  const float* vtb = vT + (size_t)bh * HDIM * SEQ;
  const float* q2b = q2 + (size_t)bh * SEQ;
  const float* k2b = k2 + (size_t)bh * SEQ;

  // Preload Q^T as B operands for all 8 K-steps of the d=32 contraction.
  v2f qB[8];
  #pragma unroll
  for (int s = 0; s < 8; ++s)
    qB[s] = *(const v2f*)(qb + (size_t)(m0 + mcol) * HDIM + s * 4 + kb);
  const float q2l = q2b[m0 + mcol];

  // Per-lane V^T row bases (rows dd = mcol and mcol+16), contiguous in j.
  const float* vrow0 = vtb + (size_t)mcol * SEQ + kb;
  const float* vrow1 = vtb + (size_t)(mcol + 16) * SEQ + kb;

  float run_max = -3.0e38f, run_sum = 0.f;
  v8f O0 = {0.f,0.f,0.f,0.f,0.f,0.f,0.f,0.f};
  v8f O1 = O0;

  for (int j0 = 0; j0 < SEQ; j0 += 16) {
    // S^T = K_tile(16x32) x Q^T(32x16)
    v8f S = {0.f,0.f,0.f,0.f,0.f,0.f,0.f,0.f};
    const float* krow = kbp + (size_t)(j0 + mcol) * HDIM + kb;
    #pragma unroll
    for (int s = 0; s < 8; ++s) {
      v2f a = *(const v2f*)(krow + s * 4);
      S = wmma4(a, qB[s], S);
    }

    // scores: -sqrt(max(q2 + k2 - 2*qk, eps)) * scale; online softmax
    const int jrbase = j0 + (hi ? 8 : 0);
    v4f k2lo = *(const v4f*)(k2b + jrbase);
    v4f k2hi = *(const v4f*)(k2b + jrbase + 4);
    float p[8];
    float tmax = -3.0e38f;
    #pragma unroll
    for (int r = 0; r < 8; ++r) {
      const float k2v = (r < 4) ? k2lo[r] : k2hi[r - 4];
      float d2 = q2l + k2v - 2.f * S[r];
      d2 = (d2 > 1e-12f) ? d2 : 1e-12f;
      float sc = -sqrtf(d2) * SCALE;
      p[r] = sc;
      tmax = (tmax > sc) ? tmax : sc;
    }
    tmax = fmaxf(tmax, __shfl_xor(tmax, 16, 32));
    const float nmax = fmaxf(run_max, tmax);
    const float corr = __expf(run_max - nmax);
    float tsum = 0.f;
    #pragma unroll
    for (int r = 0; r < 8; ++r) { p[r] = __expf(p[r] - nmax); tsum += p[r]; }
    tsum += __shfl_xor(tsum, 16, 32);
    run_sum = run_sum * corr + tsum;
    run_max = nmax;
    #pragma unroll
    for (int r = 0; r < 8; ++r) { O0[r] *= corr; O1[r] *= corr; }

    // P^T D-layout -> B-layout: half-wave swaps
    float xp[8];
    #pragma unroll
    for (int r = 0; r < 8; ++r) xp[r] = __shfl_xor(p[r], 16, 32);
    v2f bs[4];
    bs[0].x = hi ? xp[2] : p[0];  bs[0].y = hi ? xp[3] : p[1];
    bs[1].x = hi ? xp[6] : p[4];  bs[1].y = hi ? xp[7] : p[5];
    bs[2].x = hi ? p[2]  : xp[0]; bs[2].y = hi ? p[3]  : xp[1];
    bs[3].x = hi ? p[6]  : xp[4]; bs[3].y = hi ? p[7]  : xp[5];

    // O^T(32x16) += V^T(32x16key) x P^T(16key x 16query)
    #pragma unroll
    for (int s = 0; s < 4; ++s) {
      v2f a0 = *(const v2f*)(vrow0 + j0 + s * 4);
      v2f a1 = *(const v2f*)(vrow1 + j0 + s * 4);
      O0 = wmma4(a0, bs[s], O0);
      O1 = wmma4(a1, bs[s], O1);
    }
  }

  const float inv = 1.f / run_sum;
  v8f s0, s1;
  #pragma unroll
  for (int r = 0; r < 8; ++r) { s0[r] = O0[r] * inv; s1[r] = O1[r] * inv; }

  // att[b][n][h*32 + dd]: rows dd contiguous -> b128 stores
  const int b = bh >> 3, h = bh & 7;
  float* ab = att + ((size_t)b * SEQ + m0 + mcol) * INNER + h * HDIM;
  v4f* p0 = (v4f*)&s0;
  v4f* p1 = (v4f*)&s1;
  float* d0 = ab + (hi ? 8 : 0);
  float* d1 = ab + 16 + (hi ? 8 : 0);
  *(v4f*)d0       = p0[0];
  *(v4f*)(d0 + 4) = p0[1];
  *(v4f*)d1       = p1[0];
  *(v4f*)(d1 + 4) = p1[1];
}

// ---------------------------------------------------------------------------
// Kernel 4: output projection  y[b,o,n] = sum_c w_out[o,c]*att[b,n,c] + b_out
// Orientation: M = n (rows), N = o (cols) ->
//   A = att tile (contiguous b64, shared across the 4 o-tiles),
//   B = w_out^T tile (contiguous b64), D rows = n: stores contiguous (b128),
//   bias uniform per lane (one scalar load per tile).
// ---------------------------------------------------------------------------
__global__ void __launch_bounds__(256)
out_kernel(const float* __restrict__ att, const float* __restrict__ wout,
           const float* __restrict__ bout, float* __restrict__ y) {
  const int lane = threadIdx.x & 31;
  const int wave = threadIdx.x >> 5;
  const int b    = blockIdx.x >> 7;
  const int n0   = (blockIdx.x & 127) * 16;
  const int mcol = lane & 15;
  const bool hi  = lane >= 16;
  const int kb   = hi ? 2 : 0;
  const int o_base = wave * 64;
  const float* arow = att + ((size_t)b * SEQ + n0 + mcol) * INNER + kb;

  v8f acc[4];
  v8f z = {0.f,0.f,0.f,0.f,0.f,0.f,0.f,0.f};
  #pragma unroll
  for (int t = 0; t < 4; ++t) acc[t] = z;

  for (int c0 = 0; c0 < INNER; c0 += 4) {
    v2f a = *(const v2f*)(arow + c0);        // A: att (n x c)
    #pragma unroll
    for (int t = 0; t < 4; ++t) {
      const int o = o_base + t * 16 + mcol;  // B column per lane
      v2f bb = *(const v2f*)(wout + (size_t)o * INNER + c0 + kb);
      acc[t] = wmma4(a, bb, acc[t]);
    }
  }

  #pragma unroll
  for (int t = 0; t < 4; ++t) {
    const int o = o_base + t * 16 + mcol;    // D column per lane
    const float bias = bout[o];
    v8f out;
    #pragma unroll
    for (int r = 0; r < 8; ++r) out[r] = acc[t][r] + bias;
    float* p = y + ((size_t)b * DIM + o) * SEQ + n0 + (hi ? 8 : 0);
    v4f* po = (v4f*)&out;
    *(v4f*)p       = po[0];
    *(v4f*)(p + 4) = po[1];
  }
}

// ---------------------------------------------------------------------------
extern "C" void kernel_launch(void* const* d_in, const int* in_sizes, int n_in,
                              void* d_out, int out_size, void* d_ws, size_t ws_size,
                              hipStream_t stream) {
  const float* x    = (const float*)d_in[0];
  const float* wqkv = (const float*)d_in[1];
  const float* wout = (const float*)d_in[2];
  const float* bout = (const float*)d_in[3];
  float* y = (float*)d_out;

  float* ws = (float*)d_ws;
  const size_t qkvN = (size_t)NB * NHEADS * SEQ * HDIM;    // 2,097,152 floats
  float* q   = ws;
  float* k   = q  + qkvN;
  float* vT  = k  + qkvN;                                  // (B,H,d,N)
  float* q2  = vT + qkvN;
  float* k2  = q2 + (size_t)NB * NHEADS * SEQ;
  float* att = k2 + (size_t)NB * NHEADS * SEQ;             // (B,N,INNER)

  qkv_kernel <<<NB * (SEQ / 16),         256, 0, stream>>>(x, wqkv, q, k, vT);
  norm_kernel<<<(NB * NHEADS * SEQ)/256, 256, 0, stream>>>(q, k, q2, k2);
  attn_kernel<<<NB * NHEADS * (SEQ/128), 256, 0, stream>>>(q, k, vT, q2, k2, att);
  out_kernel <<<NB * (SEQ / 16),         256, 0, stream>>>(att, wout, bout, y);
}